// UniGCNIIConv_81020263071818
// MI455X (gfx1250) — compile-verified
//
#include <hip/hip_runtime.h>
#include <hip/hip_bf16.h>

#define DD 256

typedef __attribute__((ext_vector_type(2))) float v2f;
typedef __attribute__((ext_vector_type(8))) float v8f;

// degV[n] = rsqrt(sum_e H[n,e]), inf -> 1.  One wave32 per row, coalesced.
__global__ void degv_kernel(const float* __restrict__ H, float* __restrict__ degV,
                            int N, int E) {
    int wave = threadIdx.x >> 5;
    int lane = threadIdx.x & 31;
    int row  = blockIdx.x * 8 + wave;
    if (row >= N) return;
    const float* h = H + (size_t)row * E;
    float s = 0.f;
    for (int e = lane; e < E; e += 32) s += h[e];
    #pragma unroll
    for (int off = 16; off; off >>= 1) s += __shfl_xor(s, off, 32);
    if (lane == 0) degV[row] = (s > 0.f) ? rsqrtf(s) : 1.0f;
}

// Partial column sums of H accumulated with f32 atomics (colsum pre-zeroed).
__global__ void colsum_kernel(const float* __restrict__ H, float* __restrict__ colsum,
                              int N, int E, int rowsPerBlock) {
    int e = blockIdx.x * blockDim.x + threadIdx.x;
    if (e >= E) return;
    int n0 = blockIdx.y * rowsPerBlock;
    int n1 = min(n0 + rowsPerBlock, N);
    float s = 0.f;
    for (int n = n0; n < n1; ++n) s += H[(size_t)n * E + e];
    atomicAdd(&colsum[e], s);
}

// Xe[e,:] += X[vertex[j],:]; cnt[e] += 1.  64 threads (float4 slices) per nnz.
__global__ void scatter_edge_kernel(const float* __restrict__ X,
                                    const int* __restrict__ vertex,
                                    const int* __restrict__ edges,
                                    float* __restrict__ Xe, float* __restrict__ cnt,
                                    int nnz) {
    int t = blockIdx.x * blockDim.x + threadIdx.x;
    int j = t >> 6;
    int q = t & 63;
    if (j >= nnz) return;
    int v = vertex[j];
    int e = edges[j];
    float4 x = *(const float4*)(X + (size_t)v * DD + q * 4);
    float* dst = Xe + (size_t)e * DD + q * 4;
    atomicAdd(dst + 0, x.x);
    atomicAdd(dst + 1, x.y);
    atomicAdd(dst + 2, x.z);
    atomicAdd(dst + 3, x.w);
    if (q == 0) atomicAdd(&cnt[e], 1.0f);
}

// Xe[e,:] *= degE[e] / max(cnt[e],1)
__global__ void edge_scale_kernel(float* __restrict__ Xe, const float* __restrict__ cnt,
                                  const float* __restrict__ colsum, int E) {
    int t = blockIdx.x * blockDim.x + threadIdx.x;
    int e = t >> 6;
    int q = t & 63;
    if (e >= E) return;
    float s = rsqrtf(colsum[e]) / fmaxf(cnt[e], 1.0f);
    float4* p = (float4*)(Xe + (size_t)e * DD + q * 4);
    float4 x = *p;
    x.x *= s; x.y *= s; x.z *= s; x.w *= s;
    *p = x;
}

// Xv[vertex[j],:] += Xe[edges[j],:]
__global__ void scatter_vertex_kernel(const float* __restrict__ Xe,
                                      const int* __restrict__ vertex,
                                      const int* __restrict__ edges,
                                      float* __restrict__ Xv, int nnz) {
    int t = blockIdx.x * blockDim.x + threadIdx.x;
    int j = t >> 6;
    int q = t & 63;
    if (j >= nnz) return;
    int v = vertex[j];
    int e = edges[j];
    float4 x = *(const float4*)(Xe + (size_t)e * DD + q * 4);
    float* dst = Xv + (size_t)v * DD + q * 4;
    atomicAdd(dst + 0, x.x);
    atomicAdd(dst + 1, x.y);
    atomicAdd(dst + 2, x.z);
    atomicAdd(dst + 3, x.w);
}

// Xi = 0.9 * degV[n] * Xv + 0.1 * X0   (in place over Xv)
__global__ void xi_kernel(float* __restrict__ Xv, const float* __restrict__ X0,
                          const float* __restrict__ degV, int N) {
    int t = blockIdx.x * blockDim.x + threadIdx.x;
    int n = t >> 6;
    int q = t & 63;
    if (n >= N) return;
    float dv = degV[n];
    float4 xv = *(float4*)(Xv + (size_t)n * DD + q * 4);
    const float4 x0 = *(const float4*)(X0 + (size_t)n * DD + q * 4);
    float4 xi;
    xi.x = 0.9f * (xv.x * dv) + 0.1f * x0.x;
    xi.y = 0.9f * (xv.y * dv) + 0.1f * x0.y;
    xi.z = 0.9f * (xv.z * dv) + 0.1f * x0.z;
    xi.w = 0.9f * (xv.w * dv) + 0.1f * x0.w;
    *(float4*)(Xv + (size_t)n * DD + q * 4) = xi;
}

// out = 0.2*Xi + 0.8*(Xi @ W^T) via V_WMMA_F32_16X16X4_F32.
// One wave per 16x16 output tile; K=256 in steps of 4 (64 WMMAs/tile).
// A layout (16x4 f32): lane%16 = M, lane/16 selects K pair -> float2 load.
// B layout (4x16 f32): lane%16 = N; W is [out,in] row-major so B col o = W row o.
// C/D layout: VGPR r, lane -> (M = r + 8*(lane/16), N = lane%16).
__global__ void gemm_wmma_kernel(const float* __restrict__ Xi, const float* __restrict__ W,
                                 float* __restrict__ out, int M) {
    int wave = threadIdx.x >> 5;
    int lane = threadIdx.x & 31;
    int tile = blockIdx.x * 8 + wave;
    int nt = tile & 15;          // 256/16 = 16 n-tiles
    int mt = tile >> 4;
    int m0 = mt * 16, n0 = nt * 16;
    if (m0 >= M) return;
    int col  = lane & 15;
    int half = lane >> 4;
    const float* arow = Xi + (size_t)(m0 + col) * DD;  // A row M = lane%16
    const float* brow = W  + (size_t)(n0 + col) * DD;  // B col N = lane%16 == W row
    v8f acc = {};
    #pragma unroll 8
    for (int k0 = 0; k0 < DD; k0 += 4) {
        v2f a = *(const v2f*)(arow + k0 + 2 * half);
        v2f b = *(const v2f*)(brow + k0 + 2 * half);
        acc = __builtin_amdgcn_wmma_f32_16x16x4_f32(
            /*neg_a=*/false, a, /*neg_b=*/false, b,
            /*c_mod=*/(short)0, acc, /*reuse_a=*/false, /*reuse_b=*/false);
    }
    #pragma unroll
    for (int r = 0; r < 8; ++r) {
        int m = m0 + r + 8 * half;
        int n = n0 + col;
        size_t idx = (size_t)m * DD + n;
        out[idx] = 0.2f * Xi[idx] + 0.8f * acc[r];
    }
}

extern "C" void kernel_launch(void* const* d_in, const int* in_sizes, int n_in,
                              void* d_out, int out_size, void* d_ws, size_t ws_size,
                              hipStream_t stream) {
    const float* X      = (const float*)d_in[0];
    const float* X0     = (const float*)d_in[1];
    const float* H      = (const float*)d_in[2];
    const float* W      = (const float*)d_in[3];
    const int*   vertex = (const int*)d_in[4];
    const int*   edges  = (const int*)d_in[5];
    float* out = (float*)d_out;

    const int N   = in_sizes[0] / DD;        // 30000
    const int E   = in_sizes[2] / N;         // 4000
    const int NNZ = in_sizes[4];             // 960000

    // Workspace layout (256B-aligned slices)
    float* ws     = (float*)d_ws;
    float* degV   = ws;
    float* colsum = degV   + ((N + 63) & ~63);
    float* cnt    = colsum + ((E + 63) & ~63);
    float* Xe     = cnt    + ((E + 63) & ~63);
    float* Xv     = Xe     + (size_t)E * DD;   // doubles as Xi after xi_kernel

    hipMemsetAsync(colsum, 0, (size_t)E * sizeof(float), stream);
    hipMemsetAsync(cnt,    0, (size_t)E * sizeof(float), stream);
    hipMemsetAsync(Xe,     0, (size_t)E * DD * sizeof(float), stream);
    hipMemsetAsync(Xv,     0, (size_t)N * DD * sizeof(float), stream);

    // degree scalings (reads H once each: 480 MB dominant HBM traffic)
    degv_kernel<<<(N + 7) / 8, 256, 0, stream>>>(H, degV, N, E);
    {
        const int rows = 256;
        dim3 grid((E + 255) / 256, (N + rows - 1) / rows);
        colsum_kernel<<<grid, 256, 0, stream>>>(H, colsum, N, E, rows);
    }

    // vertex -> edge scatter-mean
    {
        long long threads = (long long)NNZ * 64;
        scatter_edge_kernel<<<(int)((threads + 255) / 256), 256, 0, stream>>>(
            X, vertex, edges, Xe, cnt, NNZ);
    }
    edge_scale_kernel<<<(E * 64 + 255) / 256, 256, 0, stream>>>(Xe, cnt, colsum, E);

    // edge -> vertex scatter-sum
    {
        long long threads = (long long)NNZ * 64;
        scatter_vertex_kernel<<<(int)((threads + 255) / 256), 256, 0, stream>>>(
            Xe, vertex, edges, Xv, NNZ);
    }

    // Xi = 0.9*degV*Xv + 0.1*X0  (in place)
    {
        long long threads = (long long)N * 64;
        xi_kernel<<<(int)((threads + 255) / 256), 256, 0, stream>>>(Xv, X0, degV, N);
    }

    // out = 0.2*Xi + 0.8*Xi@W^T  (f32 WMMA)
    {
        int mtiles = (N + 15) / 16;
        int tiles  = mtiles * 16;
        gemm_wmma_kernel<<<(tiles + 7) / 8, 256, 0, stream>>>(Xv, W, out, N);
    }
}